// EmbedRefine_hmvis_71932112274118
// MI455X (gfx1250) — compile-verified
//
#include <hip/hip_runtime.h>
#include <hip/hip_bf16.h>
#include <math.h>

// ---------------- problem constants (from reference) ----------------
constexpr int B_   = 8;
constexpr int D_   = 128;
constexpr int H_   = 152;
constexpr int W_   = 272;
constexpr int HW_  = H_ * W_;      // 41344
constexpr int KTOP = 500;
constexpr int NQ   = B_ * KTOP;    // 4000
constexpr int DFF_ = 512;
constexpr float EPS_ = 1e-5f;

typedef __attribute__((ext_vector_type(2))) float v2f;
typedef __attribute__((ext_vector_type(8))) float v8f;

// =====================================================================
// Wave-level GEMM: O(16 x Ncols) = scale * (A(16 x Kdim) @ W^T + bias)
//   A in LDS, row stride lda;  W global, row-major (Ncols x Kdim), row
//   stride ldw;  O in LDS, row stride ldo.  Optional ReLU.
// Uses V_WMMA_F32_16X16X4_F32 (fp32 tensor op, wave32).
// ISA lane layout:
//   A frag (16x4): lane L -> M=L&15, a[v] = K (kb + v + 2*(L>>4))
//   B frag (4x16): lane L -> N=L&15, b[v] = K (kb + v + 2*(L>>4))
//   C/D (16x16)  : lane L -> N=L&15, acc[r] = M (r + 8*(L>>4))
// =====================================================================
__device__ __forceinline__ void wave_gemm(const float* __restrict__ A, int lda,
                                          const float* __restrict__ Wg, int ldw,
                                          const float* __restrict__ bias,
                                          float* __restrict__ O, int ldo,
                                          int Ncols, int Kdim,
                                          float scale, bool relu)
{
    const int lane = threadIdx.x & 31;
    const int half = lane >> 4;      // 0 or 1
    const int l16  = lane & 15;

    for (int nb = 0; nb < Ncols; nb += 16) {
        float bval = bias ? bias[nb + l16] : 0.0f;
        v8f acc;
#pragma unroll
        for (int r = 0; r < 8; ++r) acc[r] = bval;

        const float* arow = A + l16 * lda + 2 * half;
        const float* brow = Wg + (nb + l16) * ldw + 2 * half;
        for (int kb = 0; kb < Kdim; kb += 4) {
            v2f a, b;
            a[0] = arow[kb + 0];
            a[1] = arow[kb + 1];
            b[0] = brow[kb + 0];
            b[1] = brow[kb + 1];
            acc = __builtin_amdgcn_wmma_f32_16x16x4_f32(
                /*neg_a=*/false, a, /*neg_b=*/false, b,
                /*c_mod=*/(short)0, acc, /*reuse_a=*/false, /*reuse_b=*/false);
        }
#pragma unroll
        for (int r = 0; r < 8; ++r) {
            float v = acc[r] * scale;
            if (relu) v = fmaxf(v, 0.0f);
            O[(r + 8 * half) * ldo + nb + l16] = v;
        }
    }
}

// Row-wise layernorm of (A + Badd), 16 rows x 128, lanes 0..15 own a row.
__device__ __forceinline__ void layernorm16(const float* __restrict__ A,
                                            const float* __restrict__ Badd,
                                            const float* __restrict__ g,
                                            const float* __restrict__ be,
                                            float* __restrict__ O)
{
    const int lane = threadIdx.x & 31;
    if (lane < 16) {
        const float* ar = A + lane * 128;
        const float* br = Badd + lane * 128;
        float sum = 0.0f, sq = 0.0f;
        for (int c = 0; c < 128; ++c) {
            float v = ar[c] + br[c];
            sum += v; sq += v * v;
        }
        float mu  = sum * (1.0f / 128.0f);
        float var = sq * (1.0f / 128.0f) - mu * mu;
        float inv = rsqrtf(var + EPS_);
        float* orow = O + lane * 128;
        for (int c = 0; c < 128; ++c) {
            float v = ar[c] + br[c];
            orow[c] = (v - mu) * inv * g[c] + be[c];
        }
    }
}

// =====================================================================
// Kernel 1: heat = hm*vis, 3x3 SAME max-pool NMS mask -> scratch heat
// =====================================================================
__global__ void nms_kernel(const float* __restrict__ hm,
                           const float* __restrict__ vis,
                           float* __restrict__ heat)
{
    int idx = blockIdx.x * blockDim.x + threadIdx.x;
    if (idx >= B_ * HW_) return;
    int b  = idx / HW_;
    int hw = idx - b * HW_;
    int hh = hw / W_;
    int ww = hw - hh * W_;
    const float* hb = hm  + (long)b * HW_;
    const float* vb = vis + (long)b * HW_;
    float h  = hb[hw] * vb[hw];
    float mx = -INFINITY;
#pragma unroll
    for (int dy = -1; dy <= 1; ++dy) {
#pragma unroll
        for (int dx = -1; dx <= 1; ++dx) {
            int y = hh + dy, x = ww + dx;
            if (y >= 0 && y < H_ && x >= 0 && x < W_) {
                float v = hb[y * W_ + x] * vb[y * W_ + x];
                mx = fmaxf(mx, v);
            }
        }
    }
    heat[idx] = (mx == h) ? h : 0.0f;
}

// =====================================================================
// Kernel 2: per-batch top-K by 500 iterative argmax rounds.
// Key = (monotone float bits << 32) | (HW-1-index): max key -> max value,
// ties broken toward the smallest index (matches lax.top_k).
// =====================================================================
__global__ __launch_bounds__(1024) void topk_kernel(float* __restrict__ heat,
                                                    int* __restrict__ inds)
{
    __shared__ unsigned long long red[1024];
    const int b = blockIdx.x;
    const int tid = threadIdx.x;
    float* hb = heat + (long)b * HW_;

    for (int it = 0; it < KTOP; ++it) {
        unsigned long long best = 0ull;
        for (int i = tid; i < HW_; i += 1024) {
            unsigned u = __float_as_uint(hb[i]);
            unsigned mono = (u & 0x80000000u) ? ~u : (u | 0x80000000u);
            unsigned long long key =
                ((unsigned long long)mono << 32) | (unsigned)(HW_ - 1 - i);
            if (key > best) best = key;
        }
        red[tid] = best;
        __syncthreads();
        for (int s = 512; s > 0; s >>= 1) {
            if (tid < s && red[tid + s] > red[tid]) red[tid] = red[tid + s];
            __syncthreads();
        }
        if (tid == 0) {
            int idx = HW_ - 1 - (int)(red[0] & 0xffffffffu);
            inds[b * KTOP + it] = idx;
            hb[idx] = -INFINITY;   // mask for next round
        }
        __syncthreads();
    }
}

// =====================================================================
// Kernel 3: decoder layer. One wave (32 threads) per 16-query tile.
// All GEMMs via V_WMMA_F32_16X16X4_F32; online softmax over the 9 keys.
// Writes refined features directly into `out` (already a copy of x).
// =====================================================================
__global__ __launch_bounds__(32) void decoder_kernel(
    const float* __restrict__ x, const int* __restrict__ inds,
    const float* __restrict__ in_w, const float* __restrict__ in_b,
    const float* __restrict__ ow,   const float* __restrict__ ob,
    const float* __restrict__ w1,   const float* __restrict__ b1,
    const float* __restrict__ w2,   const float* __restrict__ b2,
    const float* __restrict__ g2,   const float* __restrict__ be2,
    const float* __restrict__ g3,   const float* __restrict__ be3,
    float* __restrict__ out)
{
    // LDS layout (floats): 14336 * 4B = 56 KB (+ meta)
    __shared__ float smem[14336];
    __shared__ int s_b[16], s_ind[16];
    float* q_t  = smem + 0;      // 16x128 raw query features
    float* o_t  = smem + 2048;   // 16x128 generic GEMM output
    float* tgt  = smem + 4096;   // 16x128 post-attn LN
    float* a_t  = smem + 6144;   // 16x128 gathered neighbor tile
    float* v_t  = smem + 8192;   // 16x128 V projection
    float* qp_t = smem + 10240;  // 16x128 scaled Q projection
    float* ctx  = smem + 12288;  // 16x128 attention context accumulator
    float* ffh  = smem + 6144;   // 16x512 FFN hidden, aliases a/v/qp/ctx (dead by then)

    const int lane = threadIdx.x;
    const int t0 = blockIdx.x * 16;

    if (lane < 16) {
        int g = t0 + lane;
        int b = g / KTOP;
        s_b[lane]   = b;
        s_ind[lane] = inds[g];           // inds is laid out [b*KTOP + k] = g
        (void)b;
    }
    __syncthreads();

    // Gather q tile: x[b, c, ind] (channel stride HW)
    for (int i = lane; i < 16 * 128; i += 32) {
        int qi = i >> 7, c = i & 127;
        q_t[i] = x[(s_b[qi] * D_ + c) * HW_ + s_ind[qi]];
    }
    __syncthreads();

    // qp = (q @ Wq^T + qb) * hd^-0.5  (hd=16 -> 0.25)
    wave_gemm(q_t, 128, in_w, 128, in_b, qp_t, 128, 128, 128, 0.25f, false);
    __syncthreads();

    // online-softmax state: each lane owns 4 (query,head) pairs
    for (int i = lane; i < 2048; i += 32) ctx[i] = 0.0f;
    float m_[4], s_[4];
#pragma unroll
    for (int i = 0; i < 4; ++i) { m_[i] = -INFINITY; s_[i] = 0.0f; }

    const int offs[9] = {-W_ - 1, -W_, -W_ + 1, -1, 0, 1, W_ - 1, W_, W_ + 1};

    for (int j = 0; j < 9; ++j) {
        __syncthreads();
        // gather neighbor tile (flat-index clip, exactly as reference)
        for (int i = lane; i < 2048; i += 32) {
            int qi = i >> 7, c = i & 127;
            int nd = s_ind[qi] + offs[j];
            nd = min(max(nd, 0), HW_ - 1);
            a_t[i] = x[(s_b[qi] * D_ + c) * HW_ + nd];
        }
        __syncthreads();
        // kp_j, vp_j
        wave_gemm(a_t, 128, in_w + 128 * 128,     128, in_b + 128, o_t, 128, 128, 128, 1.0f, false);
        wave_gemm(a_t, 128, in_w + 2 * 128 * 128, 128, in_b + 256, v_t, 128, 128, 128, 1.0f, false);
        __syncthreads();
        // attention update (VALU): logit = qp . kp per head, online softmax
#pragma unroll
        for (int i = 0; i < 4; ++i) {
            int p  = lane * 4 + i;
            int qi = p >> 3, h = p & 7;
            const float* qpv = qp_t + qi * 128 + h * 16;
            const float* kv  = o_t  + qi * 128 + h * 16;
            float lg = 0.0f;
#pragma unroll
            for (int e = 0; e < 16; ++e) lg += qpv[e] * kv[e];
            float mn   = fmaxf(m_[i], lg);
            float corr = expf(m_[i] - mn);   // first iter: exp(-inf)=0
            float wgt  = expf(lg - mn);
            s_[i] = s_[i] * corr + wgt;
            float* cv = ctx + qi * 128 + h * 16;
            const float* vv = v_t + qi * 128 + h * 16;
#pragma unroll
            for (int e = 0; e < 16; ++e) cv[e] = cv[e] * corr + wgt * vv[e];
            m_[i] = mn;
        }
    }
    __syncthreads();
    // normalize context
#pragma unroll
    for (int i = 0; i < 4; ++i) {
        int p = lane * 4 + i;
        int qi = p >> 3, h = p & 7;
        float inv = 1.0f / s_[i];
        float* cv = ctx + qi * 128 + h * 16;
#pragma unroll
        for (int e = 0; e < 16; ++e) cv[e] *= inv;
    }
    __syncthreads();

    // attn_out = ctx @ Wo^T + ob
    wave_gemm(ctx, 128, ow, 128, ob, o_t, 128, 128, 128, 1.0f, false);
    __syncthreads();

    // tgt = LN(q + attn_out; g2, be2)
    layernorm16(q_t, o_t, g2, be2, tgt);
    __syncthreads();

    // FFN: ffh = relu(tgt @ w1^T + b1);  ff2 = ffh @ w2^T + b2
    wave_gemm(tgt, 128, w1, 128, b1, ffh, 512, DFF_, 128, 1.0f, true);
    __syncthreads();
    wave_gemm(ffh, 512, w2, 512, b2, o_t, 128, 128, DFF_, 1.0f, false);
    __syncthreads();

    // out = LN(tgt + ff; g3, be3) -> scatter into output tensor
    layernorm16(tgt, o_t, g3, be3, o_t);
    __syncthreads();

    for (int i = lane; i < 2048; i += 32) {
        int qi = i >> 7, c = i & 127;
        out[(s_b[qi] * D_ + c) * HW_ + s_ind[qi]] = o_t[i];
    }
}

// =====================================================================
// Launch
// =====================================================================
extern "C" void kernel_launch(void* const* d_in, const int* in_sizes, int n_in,
                              void* d_out, int out_size, void* d_ws, size_t ws_size,
                              hipStream_t stream)
{
    const float* x    = (const float*)d_in[0];
    const float* hm   = (const float*)d_in[1];
    // d_in[2]=wh, d_in[3]=reg feed only `dets`, which the reference deletes.
    const float* vis  = (const float*)d_in[4];
    const float* in_w = (const float*)d_in[5];
    const float* in_b = (const float*)d_in[6];
    const float* ow   = (const float*)d_in[7];
    const float* ob   = (const float*)d_in[8];
    const float* w1   = (const float*)d_in[9];
    const float* b1   = (const float*)d_in[10];
    const float* w2   = (const float*)d_in[11];
    const float* b2   = (const float*)d_in[12];
    const float* g2   = (const float*)d_in[13];
    const float* be2  = (const float*)d_in[14];
    const float* g3   = (const float*)d_in[15];
    const float* be3  = (const float*)d_in[16];
    float* out = (float*)d_out;

    float* heat = (float*)d_ws;                                   // B*HW floats
    int*   inds = (int*)((char*)d_ws + (size_t)B_ * HW_ * sizeof(float)); // NQ ints

    // Bulk copy x -> out (dominant traffic: ~340 MB round trip).
    hipMemcpyAsync(out, x, (size_t)B_ * D_ * HW_ * sizeof(float),
                   hipMemcpyDeviceToDevice, stream);

    // NMS mask
    int n = B_ * HW_;
    nms_kernel<<<(n + 255) / 256, 256, 0, stream>>>(hm, vis, heat);

    // Top-K per batch
    topk_kernel<<<B_, 1024, 0, stream>>>(heat, inds);

    // Decoder layer over 250 tiles of 16 queries, one wave each
    decoder_kernel<<<NQ / 16, 32, 0, stream>>>(x, inds, in_w, in_b, ow, ob,
                                               w1, b1, w2, b2, g2, be2, g3, be3,
                                               out);
}